// GCNLayer_39290360824041
// MI455X (gfx1250) — compile-verified
//
#include <hip/hip_runtime.h>
#include <hip/hip_bf16.h>

typedef __attribute__((ext_vector_type(2))) float v2f;
typedef __attribute__((ext_vector_type(8))) float v8f;

#define IN_F 256
#define OUT_F 128
#define LDS_STRIDE 264   // 256 + 8 pad floats: breaks 64-bank conflicts on column reads

// ---------------------------------------------------------------------------
// Kernel 1: h = x @ w  using V_WMMA_F32_16X16X4_F32 (exact fp32 matrix path)
// Block = 256 threads = 8 waves. Each block: 16 rows of x; wave w -> cols 16w..16w+15.
// ---------------------------------------------------------------------------
__global__ __launch_bounds__(256) void gcn_gemm_wmma(const float* __restrict__ x,
                                                     const float* __restrict__ w,
                                                     float* __restrict__ h,
                                                     int nNodes) {
    __shared__ float As[16 * LDS_STRIDE];

    const int tid  = threadIdx.x;
    const int lane = tid & 31;
    const int wave = tid >> 5;            // 0..7 : which 16-col tile of OUT_F
    const int row0 = blockIdx.x * 16;
    if (row0 >= nNodes) return;           // uniform per block

    // ---- cooperative coalesced load of A tile: 16 rows x 256 floats ----
    {
        const float4* xv = reinterpret_cast<const float4*>(x + (size_t)row0 * IN_F);
        #pragma unroll
        for (int i = 0; i < 4; ++i) {
            int f4 = tid + i * 256;       // 0..1023 (16*256/4)
            int r  = f4 >> 6;             // 64 float4 per row
            int c4 = f4 & 63;
            float4 v = xv[f4];
            float* d = &As[r * LDS_STRIDE + c4 * 4];
            d[0] = v.x; d[1] = v.y; d[2] = v.z; d[3] = v.w;
        }
    }
    __syncthreads();

    const int half = lane >> 4;           // 0: lanes 0-15, 1: lanes 16-31
    const int l15  = lane & 15;
    const int col0 = wave * 16;

    const float* arow = &As[l15 * LDS_STRIDE];  // A: M = l15
    const float* wb   = w + col0 + l15;         // B: N = col0 + l15

    v8f acc = {};
    #pragma unroll 4
    for (int k = 0; k < IN_F; k += 4) {
        const int kk = k + 2 * half;      // A lanes>=16 carry K+2,K+3 ; same for B VGPR pair
        v2f a, bm;
        a.x  = arow[kk];
        a.y  = arow[kk + 1];
        bm.x = wb[(size_t)kk * OUT_F];
        bm.y = wb[(size_t)(kk + 1) * OUT_F];
        // D = A(16x4) * B(4x16) + C   -> v_wmma_f32_16x16x4_f32
        acc = __builtin_amdgcn_wmma_f32_16x16x4_f32(
            /*neg_a=*/false, a, /*neg_b=*/false, bm,
            /*c_mod=*/(short)0, acc, /*reuse_a=*/false, /*reuse_b=*/false);
    }

    // ---- store D: VGPR v -> row = row0 + v + 8*half, col = col0 + l15 ----
    float* hb = h + (size_t)(row0 + 8 * half) * OUT_F + col0 + l15;
    #pragma unroll
    for (int v = 0; v < 8; ++v) {
        hb[(size_t)v * OUT_F] = acc[v];
    }
}

// ---------------------------------------------------------------------------
// Kernel 2: out[i][j] = b[j]   (bias pre-init so scatter accumulates on top)
// ---------------------------------------------------------------------------
__global__ void gcn_bias_init(float* __restrict__ out, const float* __restrict__ b,
                              size_t total) {
    size_t i = (size_t)blockIdx.x * blockDim.x + threadIdx.x;
    if (i < total) out[i] = b[i & (OUT_F - 1)];
}

// ---------------------------------------------------------------------------
// Kernel 3: edge scatter. One wave per edge; lane owns a float4 feature chunk.
// h fits in 192MB L2 -> gathers hit L2; atomics resolve in L2 atomic units.
// ---------------------------------------------------------------------------
__global__ __launch_bounds__(256) void gcn_scatter(const float* __restrict__ h,
                                                   const int* __restrict__ esrc,
                                                   const int* __restrict__ edst,
                                                   const float* __restrict__ ew,
                                                   float* __restrict__ out,
                                                   int nEdges) {
    const int wavesPerBlock = blockDim.x >> 5;
    const int lane   = threadIdx.x & 31;
    const int e0     = blockIdx.x * wavesPerBlock + (threadIdx.x >> 5);
    const int stride = gridDim.x * wavesPerBlock;

    for (int e = e0; e < nEdges; e += stride) {
        const int   s  = esrc[e];
        const int   d  = edst[e];
        const float wv = ew[e];

        const float4 v = reinterpret_cast<const float4*>(h + (size_t)s * OUT_F)[lane];
        float* op = out + (size_t)d * OUT_F + lane * 4;
        atomicAdd(op + 0, v.x * wv);
        atomicAdd(op + 1, v.y * wv);
        atomicAdd(op + 2, v.z * wv);
        atomicAdd(op + 3, v.w * wv);
    }
}

// ---------------------------------------------------------------------------
extern "C" void kernel_launch(void* const* d_in, const int* in_sizes, int n_in,
                              void* d_out, int out_size, void* d_ws, size_t ws_size,
                              hipStream_t stream) {
    const float* x    = (const float*)d_in[0];
    const int*   esrc = (const int*)  d_in[1];
    const int*   edst = (const int*)  d_in[2];
    const float* ew   = (const float*)d_in[3];
    const float* w    = (const float*)d_in[4];
    const float* b    = (const float*)d_in[5];
    float*       out  = (float*)d_out;
    float*       h    = (float*)d_ws;     // 100000*128*4 = 51.2 MB scratch

    const int nNodes = in_sizes[0] / IN_F;
    const int nEdges = in_sizes[3];

    // 1) dense transform with WMMA
    const int gemmBlocks = (nNodes + 15) / 16;
    gcn_gemm_wmma<<<gemmBlocks, 256, 0, stream>>>(x, w, h, nNodes);

    // 2) out = bias (broadcast)
    const size_t total = (size_t)nNodes * OUT_F;
    gcn_bias_init<<<(unsigned)((total + 255) / 256), 256, 0, stream>>>(out, b, total);

    // 3) weighted scatter-add over edges
    const int scatBlocks = (nEdges + 7) / 8;   // 8 waves (edges) per 256-thread block
    gcn_scatter<<<scatBlocks, 256, 0, stream>>>(h, esrc, edst, ew, out, nEdges);
}